// GCN2_53197464928923
// MI455X (gfx1250) — compile-verified
//
#include <hip/hip_runtime.h>
#include <hip/hip_bf16.h>
#include <math.h>

// ---------------------------------------------------------------------------
// GCN2 (GCNII) forward for MI455X / gfx1250 (wave32, WMMA)
// N=100000, E=1600000, IN=H=128, OUT=40, 8 layers
// ---------------------------------------------------------------------------

typedef float v2f __attribute__((ext_vector_type(2)));
typedef float v8f __attribute__((ext_vector_type(8)));

#define GH       128              // hidden width
#define GOUT     40               // output classes
#define GLAYERS  8
#define LDS_STRIDE 132            // A-tile row stride in floats (bank-conflict pad)

__device__ __forceinline__ void atomic_add_f32(float* p, float v) {
    // native global_atomic_add_f32 at agent scope (no CAS loop)
    __hip_atomic_fetch_add(p, v, __ATOMIC_RELAXED, __HIP_MEMORY_SCOPE_AGENT);
}

// ---------------------------------------------------------------------------
// degree / normalization
// ---------------------------------------------------------------------------
__global__ void gcn2_fill_deg(float* deg, int n) {
    int i = blockIdx.x * blockDim.x + threadIdx.x;
    if (i < n) deg[i] = 1.0f;               // self-loop
}

__global__ void gcn2_scatter_deg(const int* __restrict__ dst, float* deg, int E) {
    int i = blockIdx.x * blockDim.x + threadIdx.x;
    int stride = gridDim.x * blockDim.x;
    for (int e = i; e < E; e += stride)
        atomic_add_f32(&deg[dst[e]], 1.0f);
}

__global__ void gcn2_finalize_dinv(float* dinv, int n) {
    int i = blockIdx.x * blockDim.x + threadIdx.x;
    if (i < n) {
        float d = dinv[i];
        dinv[i] = (d > 0.0f) ? rsqrtf(d) : 0.0f;
    }
}

// ---------------------------------------------------------------------------
// mix[i] = (1-alpha)*dinv[i]^2*h[i]  +  alpha*x0[i]     (self-loop + residual)
// ---------------------------------------------------------------------------
__global__ void gcn2_seed_mix(const float* __restrict__ dinv,
                              const float* __restrict__ h,
                              const float* __restrict__ x0,
                              float* __restrict__ mix, int n) {
    int idx = blockIdx.x * blockDim.x + threadIdx.x;       // over n * 32 float4's
    if (idx >= n * (GH / 4)) return;
    int row = idx >> 5;
    float di = dinv[row];
    float w  = 0.9f * di * di;                             // (1-ALPHA) * dinv^2
    float4 hv = *(const float4*)(h  + (size_t)idx * 4);
    float4 xv = *(const float4*)(x0 + (size_t)idx * 4);
    float4 r;
    r.x = w * hv.x + 0.1f * xv.x;
    r.y = w * hv.y + 0.1f * xv.y;
    r.z = w * hv.z + 0.1f * xv.z;
    r.w = w * hv.w + 0.1f * xv.w;
    *(float4*)(mix + (size_t)idx * 4) = r;
}

// ---------------------------------------------------------------------------
// edge scatter: mix[dst] += (1-alpha)*dinv[src]*dinv[dst] * h[src]
// one wave per edge; lane handles 4 contiguous channels (float4 gather)
// ---------------------------------------------------------------------------
__global__ void gcn2_scatter_edges(const int* __restrict__ src,
                                   const int* __restrict__ dst,
                                   const float* __restrict__ dinv,
                                   const float* __restrict__ h,
                                   float* __restrict__ mix, int E) {
    int lane = threadIdx.x & 31;
    int wid  = (blockIdx.x * blockDim.x + threadIdx.x) >> 5;
    int nw   = (gridDim.x * blockDim.x) >> 5;
    for (int e = wid; e < E; e += nw) {
        int s = src[e];
        int d = dst[e];
        float w = 0.9f * dinv[s] * dinv[d];
        const float4 hv = *(const float4*)(h + (size_t)s * GH + lane * 4);
        float* p = mix + (size_t)d * GH + lane * 4;
        atomic_add_f32(p + 0, w * hv.x);
        atomic_add_f32(p + 1, w * hv.y);
        atomic_add_f32(p + 2, w * hv.z);
        atomic_add_f32(p + 3, w * hv.w);
    }
}

// ---------------------------------------------------------------------------
// WMMA GEMM (128-wide B):  C = relu?( s0*A + s1*(A@B) + bias )
// block = 256 threads (8 waves); block covers 128 rows; wave w owns col tile
// n0 = 16*w.  B fragments held in VGPRs across all 8 row tiles; A tile staged
// in LDS.  V_WMMA_F32_16X16X4_F32, K swept in steps of 4 (32 WMMAs / tile).
// ---------------------------------------------------------------------------
__global__ __launch_bounds__(256)
void gcn2_gemm128(const float* __restrict__ A, const float* __restrict__ B,
                  const float* __restrict__ bias, float s0, float s1,
                  int do_relu, float* __restrict__ C, float* __restrict__ Ccopy,
                  int nrows) {
    __shared__ float As[16 * LDS_STRIDE];
    const int tid  = threadIdx.x;
    const int wave = tid >> 5;
    const int lane = tid & 31;
    const int half = lane >> 4;
    const int lp   = lane & 15;
    const int n0   = wave * 16;
    const int rowBlock = blockIdx.x * 128;

    // B fragments: bfrag[t][v] = B[4t + 2*half + v][n0 + lp]
    v2f bfrag[32];
#pragma unroll
    for (int t = 0; t < 32; ++t) {
        int k = 4 * t + 2 * half;
        bfrag[t].x = B[(size_t)(k + 0) * GH + n0 + lp];
        bfrag[t].y = B[(size_t)(k + 1) * GH + n0 + lp];
    }
    const float bv = bias ? bias[n0 + lp] : 0.0f;

    for (int mt = 0; mt < 8; ++mt) {
        const int row0 = rowBlock + mt * 16;
        if (row0 >= nrows) break;

        __syncthreads();                      // protect previous tile's readers
        for (int idx = tid; idx < 16 * 32; idx += 256) {   // 512 float4's
            int r  = idx >> 5;
            int c4 = idx & 31;
            int rr = row0 + r;
            float4 v = make_float4(0.f, 0.f, 0.f, 0.f);
            if (rr < nrows)
                v = *(const float4*)(A + (size_t)rr * GH + c4 * 4);
            *(float4*)(&As[r * LDS_STRIDE + c4 * 4]) = v;
        }
        __syncthreads();

        v8f acc = {0.f, 0.f, 0.f, 0.f, 0.f, 0.f, 0.f, 0.f};
#pragma unroll
        for (int t = 0; t < 32; ++t) {
            int k = 4 * t + 2 * half;
            v2f a;
            a.x = As[lp * LDS_STRIDE + k + 0];
            a.y = As[lp * LDS_STRIDE + k + 1];
            acc = __builtin_amdgcn_wmma_f32_16x16x4_f32(
                false, a, false, bfrag[t], (short)0, acc, false, false);
        }

        // epilogue: D VGPR r -> (row0 + r + 8*half, n0 + lp)
        const int mbase = row0 + 8 * half;
#pragma unroll
        for (int r = 0; r < 8; ++r) {
            int m = mbase + r;
            if (m >= nrows) continue;
            float aelem = As[(8 * half + r) * LDS_STRIDE + n0 + lp];
            float v = s1 * acc[r] + s0 * aelem + bv;
            if (do_relu) v = fmaxf(v, 0.0f);
            size_t off = (size_t)m * GH + n0 + lp;
            C[off] = v;
            if (Ccopy) Ccopy[off] = v;
        }
    }
}

// ---------------------------------------------------------------------------
// Output GEMM:  logits[N,40] = h @ W1 + b1.  Cols padded to 48 (3 tiles),
// waves 0..2 compute, all 8 waves cooperate on LDS staging.
// ---------------------------------------------------------------------------
__global__ __launch_bounds__(256)
void gcn2_gemm_out(const float* __restrict__ A, const float* __restrict__ W1,
                   const float* __restrict__ b1, float* __restrict__ Out,
                   int nrows) {
    __shared__ float As[16 * LDS_STRIDE];
    const int tid  = threadIdx.x;
    const int wave = tid >> 5;
    const int lane = tid & 31;
    const int half = lane >> 4;
    const int lp   = lane & 15;
    const int n0   = wave * 16;
    const int n    = n0 + lp;                 // output column
    const bool active = (wave < 3);
    const int rowBlock = blockIdx.x * 128;

    v2f bfrag[32];
    float bv = 0.0f;
    if (active) {
#pragma unroll
        for (int t = 0; t < 32; ++t) {
            int k = 4 * t + 2 * half;
            bfrag[t].x = (n < GOUT) ? W1[(size_t)(k + 0) * GOUT + n] : 0.0f;
            bfrag[t].y = (n < GOUT) ? W1[(size_t)(k + 1) * GOUT + n] : 0.0f;
        }
        bv = (n < GOUT) ? b1[n] : 0.0f;
    }

    for (int mt = 0; mt < 8; ++mt) {
        const int row0 = rowBlock + mt * 16;
        if (row0 >= nrows) break;

        __syncthreads();
        for (int idx = tid; idx < 16 * 32; idx += 256) {
            int r  = idx >> 5;
            int c4 = idx & 31;
            int rr = row0 + r;
            float4 v = make_float4(0.f, 0.f, 0.f, 0.f);
            if (rr < nrows)
                v = *(const float4*)(A + (size_t)rr * GH + c4 * 4);
            *(float4*)(&As[r * LDS_STRIDE + c4 * 4]) = v;
        }
        __syncthreads();

        if (active) {
            v8f acc = {0.f, 0.f, 0.f, 0.f, 0.f, 0.f, 0.f, 0.f};
#pragma unroll
            for (int t = 0; t < 32; ++t) {
                int k = 4 * t + 2 * half;
                v2f a;
                a.x = As[lp * LDS_STRIDE + k + 0];
                a.y = As[lp * LDS_STRIDE + k + 1];
                acc = __builtin_amdgcn_wmma_f32_16x16x4_f32(
                    false, a, false, bfrag[t], (short)0, acc, false, false);
            }
            const int mbase = row0 + 8 * half;
#pragma unroll
            for (int r = 0; r < 8; ++r) {
                int m = mbase + r;
                if (m < nrows && n < GOUT)
                    Out[(size_t)m * GOUT + n] = acc[r] + bv;
            }
        }
    }
}

// ---------------------------------------------------------------------------
// in-place log_softmax over rows of [N,40]; one wave per row
// ---------------------------------------------------------------------------
__global__ void gcn2_log_softmax(float* __restrict__ out, int nrows) {
    int lane = threadIdx.x & 31;
    int wid  = (blockIdx.x * blockDim.x + threadIdx.x) >> 5;
    int nw   = (gridDim.x * blockDim.x) >> 5;
    for (int row = wid; row < nrows; row += nw) {
        float* r = out + (size_t)row * GOUT;
        float a = r[lane];
        float b = (lane < GOUT - 32) ? r[32 + lane] : -3.402823466e38f;
        float m = fmaxf(a, b);
        for (int s = 16; s > 0; s >>= 1) m = fmaxf(m, __shfl_xor(m, s, 32));
        float e = expf(a - m) + ((lane < GOUT - 32) ? expf(b - m) : 0.0f);
        for (int s = 16; s > 0; s >>= 1) e += __shfl_xor(e, s, 32);
        float lse = logf(e) + m;
        r[lane] = a - lse;
        if (lane < GOUT - 32) r[32 + lane] = b - lse;
    }
}

// ---------------------------------------------------------------------------
// host launcher
// ---------------------------------------------------------------------------
extern "C" void kernel_launch(void* const* d_in, const int* in_sizes, int n_in,
                              void* d_out, int out_size, void* d_ws, size_t ws_size,
                              hipStream_t stream) {
    const float* x    = (const float*)d_in[0];   // [N,128]
    const int*   ei   = (const int*)  d_in[1];   // [2,E]
    const float* W0   = (const float*)d_in[2];   // [128,128]
    const float* b0   = (const float*)d_in[3];   // [128]
    const float* Wc   = (const float*)d_in[4];   // [8,128,128]
    const float* W1   = (const float*)d_in[5];   // [128,40]
    const float* b1   = (const float*)d_in[6];   // [40]
    float* out = (float*)d_out;                  // [N,40]

    const int N = in_sizes[0] / GH;
    const int E = in_sizes[1] / 2;
    const int* e_src = ei;
    const int* e_dst = ei + E;

    // workspace layout (floats)
    size_t NA = ((size_t)N + 63) & ~(size_t)63;  // 256B-aligned node array
    float* ws   = (float*)d_ws;
    float* dinv = ws;                            // [N]   (deg, then rsqrt)
    float* h    = dinv + NA;                     // [N,128]
    float* x0   = h + (size_t)N * GH;            // [N,128]
    float* mix  = x0 + (size_t)N * GH;           // [N,128]

    const int T = 256;
    dim3 blk(T);
    int gN      = (N + T - 1) / T;
    int gNH4    = (N * (GH / 4) + T - 1) / T;
    int gGemm   = (N + 127) / 128;
    int gEdges  = 8192;                          // grid-stride over E waves
    int gSmax   = (N + 7) / 8;                   // wave per row

    // ---- normalization: deg = 1 + in-degree ; dinv = rsqrt(deg)
    gcn2_fill_deg<<<gN, blk, 0, stream>>>(dinv, N);
    gcn2_scatter_deg<<<4096, blk, 0, stream>>>(e_dst, dinv, E);
    gcn2_finalize_dinv<<<gN, blk, 0, stream>>>(dinv, N);

    // ---- input layer: h = relu(x @ W0 + b0); x0 = h
    gcn2_gemm128<<<gGemm, blk, 0, stream>>>(x, W0, b0, 0.0f, 1.0f, 1, h, x0, N);

    // ---- GCN2 layers
    for (int l = 0; l < GLAYERS; ++l) {
        float beta = logf(0.5f / (float)(l + 1) + 1.0f);
        gcn2_seed_mix<<<gNH4, blk, 0, stream>>>(dinv, h, x0, mix, N);
        gcn2_scatter_edges<<<gEdges, blk, 0, stream>>>(e_src, e_dst, dinv, h, mix, E);
        gcn2_gemm128<<<gGemm, blk, 0, stream>>>(mix, Wc + (size_t)l * GH * GH,
                                                nullptr, 1.0f - beta, beta, 1,
                                                h, nullptr, N);
    }

    // ---- output head + log_softmax
    gcn2_gemm_out<<<gGemm, blk, 0, stream>>>(h, W1, b1, out, N);
    gcn2_log_softmax<<<gSmax, blk, 0, stream>>>(out, N);
}